// GraphNeuralNetwork_44985487458708
// MI455X (gfx1250) — compile-verified
//
#include <hip/hip_runtime.h>
#include <hip/hip_bf16.h>

#define NN 100000      // N_NODES
#define NE 1600000     // N_EDGES
#define FIN 11
#define HD 128
#define DOUT 64
#define LN_EPS 1e-5f

typedef float v2f __attribute__((ext_vector_type(2)));
typedef float v8f __attribute__((ext_vector_type(8)));

// ---------------- utility: zero fill ----------------
__global__ void gnn_zero(float* __restrict__ p, int n) {
  int i = blockIdx.x * blockDim.x + threadIdx.x;
  if (i < n) p[i] = 0.0f;
}

// ---------------- degree (once) ----------------
__global__ void gnn_deg(const int* __restrict__ tgt, float* __restrict__ deg) {
  int e = blockIdx.x * blockDim.x + threadIdx.x;
  if (e < NE) unsafeAtomicAdd(&deg[tgt[e]], 1.0f);
}

// ---------------- layer-0 linear: [N,11] x W0^T[11,128] + b0 ----------------
// K=11 is too small for WMMA; one block per row, one thread per output col.
__global__ void gnn_gemm0(const float* __restrict__ x, const float* __restrict__ W0,
                          const float* __restrict__ b0, float* __restrict__ out) {
  int i = blockIdx.x;      // row
  int j = threadIdx.x;     // col 0..127
  const float* xr = x + i * FIN;
  const float* wr = W0 + j * FIN;
  float s = b0[j];
#pragma unroll
  for (int k = 0; k < FIN; ++k) s = fmaf(xr[k], wr[k], s);
  out[i * HD + j] = s;
}

// ---------------- WMMA fp32 GEMM: out[N,NC] = X[N,128] @ W^T + b ----------------
// W is [NC,128] row-major (torch Linear). One wave owns a 16-row tile and all
// NC output columns (NT = NC/16 accumulator tiles). Uses V_WMMA_F32_16X16X4_F32.
// B fragments are staged into a register array per k-step so the compiler can
// issue all NT+1 loads before a single wait, then run NT WMMAs back-to-back.
template <int NT>
__global__ void gnn_gemm_wmma(const float* __restrict__ X, const float* __restrict__ W,
                              const float* __restrict__ bias, float* __restrict__ out) {
  constexpr int NC = NT * 16;
  int wave = threadIdx.x >> 5;
  int tile = blockIdx.x * (blockDim.x >> 5) + wave;
  if (tile * 16 >= NN) return;            // whole-wave exit: EXEC stays all-ones
  int lane = threadIdx.x & 31;
  int n    = lane & 15;                   // M for A-frag, N for B/C-frags
  int kh   = (lane >> 4) << 1;            // K sub-offset: lanes0-15 -> {0,1}, 16-31 -> {2,3}
  int hi   = lane >> 4;

  const float* xrow = X + (size_t)(tile * 16 + n) * HD;
  const float* wrow = W + (size_t)n * HD;   // column n's weight row; tiles offset by 16*HD

  v8f acc[NT];
#pragma unroll
  for (int t = 0; t < NT; ++t) {
    float bv = bias[t * 16 + n];          // bias depends only on column
#pragma unroll
    for (int v = 0; v < 8; ++v) acc[t][v] = bv;
  }

#pragma unroll 2
  for (int k = 0; k < HD; k += 4) {
    v2f a = *(const v2f*)(xrow + k + kh);                     // A: 16x4 tile
    v2f bf[NT];
#pragma unroll
    for (int t = 0; t < NT; ++t)
      bf[t] = *(const v2f*)(wrow + (size_t)t * 16 * HD + k + kh);  // B: 4x16 (= W^T tile)
#pragma unroll
    for (int t = 0; t < NT; ++t)
      acc[t] = __builtin_amdgcn_wmma_f32_16x16x4_f32(
          /*neg_a=*/false, a, /*neg_b=*/false, bf[t],
          /*c_mod=*/(short)0, acc[t], /*reuse_a=*/false, /*reuse_b=*/false);
  }

  int rbase = tile * 16 + hi * 8;         // C/D layout: vgpr v -> row v (+8 for hi half)
#pragma unroll
  for (int t = 0; t < NT; ++t)
#pragma unroll
    for (int v = 0; v < 8; ++v)
      out[(size_t)(rbase + v) * NC + t * 16 + n] = acc[t][v];
}

// ---------------- edge scatter-add: one wave per edge ----------------
template <int D>   // 128 -> 4 floats/lane, 64 -> 2 floats/lane
__global__ void gnn_scatter(const float* __restrict__ xt, const int* __restrict__ src,
                            const int* __restrict__ tgt, float* __restrict__ agg) {
  constexpr int PL = D / 32;
  int gid = blockIdx.x * blockDim.x + threadIdx.x;
  int e = gid >> 5;
  if (e >= NE) return;
  int lane = threadIdx.x & 31;
  int s = src[e], t = tgt[e];
  const float* xp = xt  + (size_t)s * D + lane * PL;
  float*       ap = agg + (size_t)t * D + lane * PL;
  float ld[PL];
#pragma unroll
  for (int j = 0; j < PL; ++j) ld[j] = xp[j];   // vectorizes to b128/b64 load
#pragma unroll
  for (int j = 0; j < PL; ++j) unsafeAtomicAdd(&ap[j], ld[j]);
}

// ---------------- mean-normalize + LayerNorm (+ReLU): one wave per row ----------------
template <int D, bool RELU>
__global__ void gnn_ln(const float* __restrict__ agg, const float* __restrict__ deg,
                       const float* __restrict__ g, const float* __restrict__ be,
                       float* __restrict__ out) {
  constexpr int PL = D / 32;
  int wave = threadIdx.x >> 5;
  int row = blockIdx.x * (blockDim.x >> 5) + wave;
  if (row >= NN) return;
  int lane = threadIdx.x & 31;

  float d = deg[row];
  float inv = 1.0f / (d == 0.0f ? 1.0f : d);

  float v[PL];
  float s = 0.0f;
#pragma unroll
  for (int j = 0; j < PL; ++j) {
    v[j] = agg[(size_t)row * D + lane * PL + j] * inv;
    s += v[j];
  }
#pragma unroll
  for (int off = 16; off > 0; off >>= 1) s += __shfl_xor(s, off, 32);
  float mu = s * (1.0f / D);

  float q = 0.0f;
#pragma unroll
  for (int j = 0; j < PL; ++j) { float dd = v[j] - mu; q += dd * dd; }
#pragma unroll
  for (int off = 16; off > 0; off >>= 1) q += __shfl_xor(q, off, 32);
  float rstd = rsqrtf(q * (1.0f / D) + LN_EPS);

#pragma unroll
  for (int j = 0; j < PL; ++j) {
    int c = lane * PL + j;
    float y = (v[j] - mu) * rstd * g[c] + be[c];
    if (RELU) y = fmaxf(y, 0.0f);
    out[(size_t)row * D + c] = y;
  }
}

// ---------------- orchestration ----------------
extern "C" void kernel_launch(void* const* d_in, const int* in_sizes, int n_in,
                              void* d_out, int out_size, void* d_ws, size_t ws_size,
                              hipStream_t stream) {
  const float* x   = (const float*)d_in[0];
  const int*   ei  = (const int*)d_in[1];
  const int*   src = ei;            // edge_index[0]
  const int*   tgt = ei + NE;       // edge_index[1]
  const float* W0  = (const float*)d_in[2];
  const float* b0  = (const float*)d_in[3];
  const float* W1  = (const float*)d_in[4];
  const float* b1  = (const float*)d_in[5];
  const float* W2  = (const float*)d_in[6];
  const float* b2  = (const float*)d_in[7];
  const float* g0  = (const float*)d_in[8];
  const float* be0 = (const float*)d_in[9];
  const float* g1  = (const float*)d_in[10];
  const float* be1 = (const float*)d_in[11];
  const float* g2  = (const float*)d_in[12];
  const float* be2 = (const float*)d_in[13];
  float* out = (float*)d_out;

  // scratch layout: two N*128 ping-pong buffers + degree vector (~103 MB)
  float* bufA = (float*)d_ws;
  float* bufB = bufA + (size_t)NN * HD;
  float* deg  = bufB + (size_t)NN * HD;

  const int Z = 256;
  const int nH  = NN * HD;                 // 12.8M
  const int nO  = NN * DOUT;               // 6.4M
  const int scB = (NE * 32) / Z;           // scatter blocks (exact: 200000)
  const int lnB = NN / 8;                  // 8 waves/block, one row per wave
  const int gmB = (NN / 16 + 7) / 8;       // 16-row tiles, 8 waves/block

  // degree (shared by all layers)
  gnn_zero<<<(NN + Z - 1) / Z, Z, 0, stream>>>(deg, NN);
  gnn_deg<<<(NE + Z - 1) / Z, Z, 0, stream>>>(tgt, deg);

  // ---- layer 0: x -> bufA (linear) -> bufB (scatter) -> bufA (LN+ReLU)
  gnn_gemm0<<<NN, HD, 0, stream>>>(x, W0, b0, bufA);
  gnn_zero<<<(nH + Z - 1) / Z, Z, 0, stream>>>(bufB, nH);
  gnn_scatter<HD><<<scB, Z, 0, stream>>>(bufA, src, tgt, bufB);
  gnn_ln<HD, true><<<lnB, Z, 0, stream>>>(bufB, deg, g0, be0, bufA);

  // ---- layer 1: bufA -> bufB (WMMA gemm) -> bufA (scatter) -> bufB (LN+ReLU)
  gnn_gemm_wmma<8><<<gmB, Z, 0, stream>>>(bufA, W1, b1, bufB);
  gnn_zero<<<(nH + Z - 1) / Z, Z, 0, stream>>>(bufA, nH);
  gnn_scatter<HD><<<scB, Z, 0, stream>>>(bufB, src, tgt, bufA);
  gnn_ln<HD, true><<<lnB, Z, 0, stream>>>(bufA, deg, g1, be1, bufB);

  // ---- layer 2: bufB -> bufA[N,64] (WMMA gemm) -> bufB (scatter) -> d_out (LN)
  gnn_gemm_wmma<4><<<gmB, Z, 0, stream>>>(bufB, W2, b2, bufA);
  gnn_zero<<<(nO + Z - 1) / Z, Z, 0, stream>>>(bufB, nO);
  gnn_scatter<DOUT><<<scB, Z, 0, stream>>>(bufA, src, tgt, bufB);
  gnn_ln<DOUT, false><<<lnB, Z, 0, stream>>>(bufB, deg, g2, be2, out);
}